// Mel2LPC_38920993636627
// MI455X (gfx1250) — compile-verified
//
#include <hip/hip_runtime.h>

typedef __attribute__((ext_vector_type(2))) float v2f;
typedef __attribute__((ext_vector_type(8))) float v8f;

#define T_FRAMES 16384
#define NMELS    128
#define REPEAT   256

__global__ __launch_bounds__(256) void mel2lpc_kernel(
    const float* __restrict__ mel,    // [128, 16384]
    const float* __restrict__ basis,  // [513, 128]
    const float* __restrict__ lagw,   // [5]
    float* __restrict__ out)          // [4, 16384*256]
{
    __shared__ __align__(16) float sB[16 * 130];  // 10^mel transposed [t][m], row pad 2
    __shared__ float sAC[5 * 16];
    __shared__ float sLPC[4 * 16];

    const int tid  = threadIdx.x;
    const int lane = tid & 31;
    const int wv   = tid >> 5;
    const int tcol = lane & 15;   // N column (time) this lane owns
    const int hi   = lane >> 4;   // 0: K 0..1 / M 0..7 ; 1: K 2..3 / M 8..15
    const int t0   = blockIdx.x * 16;

    // ---- stage B: mel_lin = 10^mel, transposed into LDS ----
    {
        const int m  = tid >> 1;        // 0..127
        const int th = (tid & 1) * 8;   // 0 or 8
        const float* mp = mel + m * T_FRAMES + t0 + th;
        float4 x0 = *(const float4*)mp;
        float4 x1 = *(const float4*)(mp + 4);
        const float LN10 = 2.302585093f;
        sB[(th + 0) * 130 + m] = __expf(x0.x * LN10);
        sB[(th + 1) * 130 + m] = __expf(x0.y * LN10);
        sB[(th + 2) * 130 + m] = __expf(x0.z * LN10);
        sB[(th + 3) * 130 + m] = __expf(x0.w * LN10);
        sB[(th + 4) * 130 + m] = __expf(x1.x * LN10);
        sB[(th + 5) * 130 + m] = __expf(x1.y * LN10);
        sB[(th + 6) * 130 + m] = __expf(x1.z * LN10);
        sB[(th + 7) * 130 + m] = __expf(x1.w * LN10);
    }
    if (tid < 80) sAC[tid] = 0.0f;
    __syncthreads();

    // ---- preload the 32 B fragments (K-step = 4) for this lane ----
    v2f bf[32];
#pragma unroll
    for (int kk = 0; kk < 32; ++kk)
        bf[kk] = *(const v2f*)(&sB[tcol * 130 + kk * 4 + hi * 2]);

    float acc0 = 0.f, acc1 = 0.f, acc2 = 0.f, acc3 = 0.f, acc4 = 0.f;

    // clamp/square + cosine projection of one C tile into autocorr partials
    auto accumAC = [&](const v8f& c, int fbase) {
#pragma unroll
        for (int j = 0; j < 8; ++j) {
            const int fi = fbase + j;
            float lin = fmaxf(c[j], 1e-12f);
            float p   = lin * lin;
            float mult = (fi == 0 || fi == 512) ? (1.0f / 1024.0f) : (2.0f / 1024.0f);
            if (fi > 512) mult = 0.0f;               // padded rows
            float v   = p * mult;
            float ang = 0.006135923151543f * (float)fi;  // 2*pi/1024
            acc0 += v;
            acc1 += v * __cosf(ang);
            acc2 += v * __cosf(2.0f * ang);
            acc3 += v * __cosf(3.0f * ang);
            acc4 += v * __cosf(4.0f * ang);
        }
    };

    // ---- main GEMM: 4 full tiles per wave (rows f <= 511, no masking) ----
    for (int mt = wv; mt < 32; mt += 8) {
        const float* ap = basis + (mt * 16 + tcol) * NMELS + hi * 2;

        v2f a[32];
#pragma unroll
        for (int kk = 0; kk < 32; ++kk)        // 32 loads in flight, one wait
            a[kk] = *(const v2f*)(ap + kk * 4);

        v8f c = {0.f, 0.f, 0.f, 0.f, 0.f, 0.f, 0.f, 0.f};
#pragma unroll
        for (int kk = 0; kk < 32; ++kk)        // back-to-back WMMA chain
            c = __builtin_amdgcn_wmma_f32_16x16x4_f32(
                    false, a[kk], false, bf[kk], (short)0, c, false, false);

        accumAC(c, mt * 16 + hi * 8);
    }

    // ---- tail tile mt=32 (rows 512..527, masked), wave-uniform branch ----
    if (wv == 0) {
        const int   f   = 512 + tcol;
        const int   fcl = f > 512 ? 512 : f;
        const float msk = (f <= 512) ? 1.0f : 0.0f;
        const float* ap = basis + fcl * NMELS + hi * 2;

        v2f a[32];
#pragma unroll
        for (int kk = 0; kk < 32; ++kk) {
            v2f t = *(const v2f*)(ap + kk * 4);
            t.x *= msk;
            t.y *= msk;
            a[kk] = t;
        }

        v8f c = {0.f, 0.f, 0.f, 0.f, 0.f, 0.f, 0.f, 0.f};
#pragma unroll
        for (int kk = 0; kk < 32; ++kk)
            c = __builtin_amdgcn_wmma_f32_16x16x4_f32(
                    false, a[kk], false, bf[kk], (short)0, c, false, false);

        accumAC(c, 512 + hi * 8);
    }

    atomicAdd(&sAC[0 * 16 + tcol], acc0);
    atomicAdd(&sAC[1 * 16 + tcol], acc1);
    atomicAdd(&sAC[2 * 16 + tcol], acc2);
    atomicAdd(&sAC[3 * 16 + tcol], acc3);
    atomicAdd(&sAC[4 * 16 + tcol], acc4);
    __syncthreads();

    // ---- lag window + order-4 Levinson-Durbin (one thread per column) ----
    if (tid < 16) {
        const int t = tid;
        float A0 = sAC[0 * 16 + t] * lagw[0];
        float A1 = sAC[1 * 16 + t] * lagw[1];
        float A2 = sAC[2 * 16 + t] * lagw[2];
        float A3 = sAC[3 * 16 + t] * lagw[3];
        float A4 = sAC[4 * 16 + t] * lagw[4];

        float E = A0;
        float k = A1 / E;
        E *= fmaxf(1.0f - k * k, 1e-5f);
        float l0 = -k;

        float acc = A2 + l0 * A1;
        k = acc / E;
        E *= fmaxf(1.0f - k * k, 1e-5f);
        float n0 = l0 - k * l0;
        float l1 = -k;
        l0 = n0;

        acc = A3 + l0 * A2 + l1 * A1;
        k = acc / E;
        E *= fmaxf(1.0f - k * k, 1e-5f);
        n0 = l0 - k * l1;
        float n1 = l1 - k * l0;
        float l2 = -k;
        l0 = n0; l1 = n1;

        acc = A4 + l0 * A3 + l1 * A2 + l2 * A1;
        k = acc / E;
        n0 = l0 - k * l2;
        n1 = l1 - k * l1;
        float n2 = l2 - k * l0;
        float l3 = -k;
        l0 = n0; l1 = n1; l2 = n2;

        // lpc_ctrl = -flip(lpc): out row j = -lpc[3-j]
        sLPC[0 * 16 + t] = -l3;
        sLPC[1 * 16 + t] = -l2;
        sLPC[2 * 16 + t] = -l1;
        sLPC[3 * 16 + t] = -l0;
    }
    __syncthreads();

    // ---- write output, repeated x256, coalesced float4 stores ----
    const size_t TOUT = (size_t)T_FRAMES * REPEAT;  // 4194304
#pragma unroll
    for (int i = 0; i < 16; ++i) {
        int idx = tid + i * 256;     // 0..4095 float4 stores per block
        int j   = idx >> 10;         // lpc row 0..3
        int rem = idx & 1023;
        int tt  = rem >> 6;          // local time column 0..15
        int ch  = rem & 63;          // float4 chunk within 256 repeats
        float v = sLPC[j * 16 + tt];
        float4 vv = make_float4(v, v, v, v);
        *(float4*)(out + (size_t)j * TOUT + (size_t)(t0 + tt) * REPEAT + ch * 4) = vv;
    }
}

extern "C" void kernel_launch(void* const* d_in, const int* in_sizes, int n_in,
                              void* d_out, int out_size, void* d_ws, size_t ws_size,
                              hipStream_t stream) {
    (void)in_sizes; (void)n_in; (void)out_size; (void)d_ws; (void)ws_size;
    const float* mel   = (const float*)d_in[0];
    const float* basis = (const float*)d_in[1];
    const float* lagw  = (const float*)d_in[2];
    float* out = (float*)d_out;
    mel2lpc_kernel<<<dim3(T_FRAMES / 16), dim3(256), 0, stream>>>(mel, basis, lagw, out);
}